// GCN_Encoder_66898410602721
// MI455X (gfx1250) — compile-verified
//
#include <hip/hip_runtime.h>
#include <hip/hip_bf16.h>

typedef __attribute__((ext_vector_type(2))) float v2f;
typedef __attribute__((ext_vector_type(8))) float v8f;

// ---------------------------------------------------------------------------
// Degree / normalization kernels
// ---------------------------------------------------------------------------
__global__ void k_fill1(float* __restrict__ p, int n) {
    int i = blockIdx.x * blockDim.x + threadIdx.x;
    if (i < n) p[i] = 1.0f;   // self-loop contributes 1 to every degree
}

__global__ void k_deg(const int* __restrict__ dst, float* __restrict__ deg, int e) {
    int i = blockIdx.x * blockDim.x + threadIdx.x;
    if (i < e) atomicAdd(&deg[dst[i]], 1.0f);
}

__global__ void k_rsqrt(float* __restrict__ p, int n) {
    int i = blockIdx.x * blockDim.x + threadIdx.x;
    if (i < n) p[i] = rsqrtf(p[i]);   // deg >= 1 always (self-loops)
}

__global__ void k_norm(const int* __restrict__ src, const int* __restrict__ dst,
                       const float* __restrict__ dinv, float* __restrict__ nrm, int e) {
    int i = blockIdx.x * blockDim.x + threadIdx.x;
    if (i < e) nrm[i] = dinv[src[i]] * dinv[dst[i]];
}

// ---------------------------------------------------------------------------
// Aggregation: z = sum_{e: dst=i} norm[e]*h[src[e]] + dinv[i]^2 * h[i]
// ---------------------------------------------------------------------------
template <int D>
__global__ void k_selfloop(const float* __restrict__ h, const float* __restrict__ dinv,
                           float* __restrict__ z, int n) {
    int i = blockIdx.x * blockDim.x + threadIdx.x;
    if (i >= n * D) return;
    int node = i / D;
    float s = dinv[node];
    z[i] = h[i] * s * s;
}

template <int D>
__global__ void k_scatter(const float* __restrict__ h, const int* __restrict__ src,
                          const int* __restrict__ dst, const float* __restrict__ nrm,
                          float* __restrict__ z, int e) {
    size_t t = (size_t)blockIdx.x * blockDim.x + threadIdx.x;
    if (t >= (size_t)e * D) return;
    int edge = (int)(t / D);        // D is 64/128 -> shift
    int f    = (int)(t & (D - 1));  // consecutive lanes cover one edge row (coalesced)
    atomicAdd(&z[(size_t)dst[edge] * D + f], h[(size_t)src[edge] * D + f] * nrm[edge]);
}

template <int D>
__global__ void k_bias_relu(float* __restrict__ z, const float* __restrict__ b, int n) {
    int i = blockIdx.x * blockDim.x + threadIdx.x;
    if (i >= n * D) return;
    z[i] = fmaxf(z[i] + b[i & (D - 1)], 0.0f);
}

// ---------------------------------------------------------------------------
// WMMA f32 GEMM:  Y[nrows,DOUT] = X[nrows,DIN] @ W  (optionally W^T, +X residual, ReLU)
// One 16x16 output tile per wave32, 8 waves/block. B column-tile staged in LDS.
// Uses V_WMMA_F32_16X16X4_F32 (K=4 per issue, f32 end-to-end).
// ---------------------------------------------------------------------------
template <int DIN, int DOUT, bool TRANS_B, bool RESIDUAL, bool RELU>
__global__ __launch_bounds__(256) void gemm16(const float* __restrict__ X,
                                              const float* __restrict__ W,
                                              float* __restrict__ Y, int nrows) {
    __shared__ float Bs[DIN * 16];
    const int col0 = blockIdx.x * 16;

    // Stage the 16-wide weight column tile (row-major [DIN][16]) in LDS.
    for (int i = threadIdx.x; i < DIN * 16; i += 256) {
        const int k = i >> 4, n = i & 15;
        Bs[i] = TRANS_B ? W[(size_t)(col0 + n) * DIN + k]
                        : W[(size_t)k * DOUT + col0 + n];
    }
    __syncthreads();

    const int wave = threadIdx.x >> 5;
    const int lane = threadIdx.x & 31;
    const int rowTile = blockIdx.y * 8 + wave;
    if (rowTile * 16 >= nrows) return;        // wave-uniform: EXEC all-ones for WMMA

    const int m  = lane & 15;                 // A: M index; B: N index
    const int kh = (lane >> 4) << 1;          // K sub-pair: lanes 0-15 -> K{0,1}, 16-31 -> K{2,3}

    const float* __restrict__ xrow = X + (size_t)(rowTile * 16 + m) * DIN;

    v8f acc = {};
#pragma unroll
    for (int k = 0; k < DIN; k += 4) {
        v2f a, b;
        a.x = xrow[k + kh];
        a.y = xrow[k + kh + 1];
        b.x = Bs[(k + kh) * 16 + m];
        b.y = Bs[(k + kh + 1) * 16 + m];
        acc = __builtin_amdgcn_wmma_f32_16x16x4_f32(
            /*neg_a=*/false, a, /*neg_b=*/false, b,
            /*c_mod=*/(short)0, acc, /*reuse_a=*/false, /*reuse_b=*/false);
    }

    // C/D layout: VGPR r, lanes 0-15 -> M=r, lanes 16-31 -> M=r+8; N = lane&15.
#pragma unroll
    for (int r = 0; r < 8; ++r) {
        const int row = rowTile * 16 + r + ((lane >> 4) << 3);
        const int col = col0 + m;
        float v = acc[r];
        if (RESIDUAL) v += X[(size_t)row * DIN + col];   // residual only when DIN==DOUT
        if (RELU)     v = fmaxf(v, 0.0f);
        Y[(size_t)row * DOUT + col] = v;
    }
}

// ---------------------------------------------------------------------------
// Orchestration
// ---------------------------------------------------------------------------
extern "C" void kernel_launch(void* const* d_in, const int* in_sizes, int n_in,
                              void* d_out, int out_size, void* d_ws, size_t ws_size,
                              hipStream_t stream) {
    const int IN = 128, H1 = 128, H2 = 64, OUT = 128;
    const int N = in_sizes[0] / IN;        // 100000 (multiple of 16)
    const int E = in_sizes[1] / 2;         // 1600000

    const float* x      = (const float*)d_in[0];
    const int*   ei     = (const int*)d_in[1];
    const float* gcn1_w = (const float*)d_in[2];
    const float* gcn1_b = (const float*)d_in[3];
    const float* fc2_w  = (const float*)d_in[4];
    const float* gcn3_w = (const float*)d_in[5];
    const float* gcn3_b = (const float*)d_in[6];
    const float* fc4_w  = (const float*)d_in[7];
    const float* gcn5_w = (const float*)d_in[8];
    const float* gcn5_b = (const float*)d_in[9];
    const float* fc6_w  = (const float*)d_in[10];
    float* out = (float*)d_out;

    const int* src = ei;        // edge_index[0]
    const int* dst = ei + E;    // edge_index[1]

    float* ws   = (float*)d_ws;
    float* dinv = ws;                         // N
    float* nrm  = dinv + N;                   // E
    float* bufA = nrm + E;                    // N*128
    float* bufB = bufA + (size_t)N * 128;     // N*128

    const int T = 256;
    auto cdiv = [](long long a, long long b) { return (int)((a + b - 1) / b); };

    const int rowTiles = (N + 15) / 16;               // 6250
    const dim3 blk(T);
    const dim3 g128(H1 / 16, cdiv(rowTiles, 8));      // 8 x 782
    const dim3 g64 (H2 / 16, cdiv(rowTiles, 8));      // 4 x 782

    // --- GCN normalization: deg -> dinv -> per-edge norm ---
    k_fill1<<<cdiv(N, T), blk, 0, stream>>>(dinv, N);
    k_deg  <<<cdiv(E, T), blk, 0, stream>>>(dst, dinv, E);
    k_rsqrt<<<cdiv(N, T), blk, 0, stream>>>(dinv, N);
    k_norm <<<cdiv(E, T), blk, 0, stream>>>(src, dst, dinv, nrm, E);

    // --- Layer 1: z1 = relu(agg(x @ W1) + b1); z1 = relu(z1 + z1 @ fc2_w^T) ---
    gemm16<128, 128, false, false, false><<<g128, blk, 0, stream>>>(x, gcn1_w, bufA, N);
    k_selfloop<128><<<cdiv((long long)N * 128, T), blk, 0, stream>>>(bufA, dinv, bufB, N);
    k_scatter <128><<<cdiv((long long)E * 128, T), blk, 0, stream>>>(bufA, src, dst, nrm, bufB, E);
    k_bias_relu<128><<<cdiv((long long)N * 128, T), blk, 0, stream>>>(bufB, gcn1_b, N);
    gemm16<128, 128, true, true, true><<<g128, blk, 0, stream>>>(bufB, fc2_w, bufA, N);

    // --- Layer 2 (128 -> 64) ---
    gemm16<128, 64, false, false, false><<<g64, blk, 0, stream>>>(bufA, gcn3_w, bufB, N);
    k_selfloop<64><<<cdiv((long long)N * 64, T), blk, 0, stream>>>(bufB, dinv, bufA, N);
    k_scatter <64><<<cdiv((long long)E * 64, T), blk, 0, stream>>>(bufB, src, dst, nrm, bufA, E);
    k_bias_relu<64><<<cdiv((long long)N * 64, T), blk, 0, stream>>>(bufA, gcn3_b, N);
    gemm16<64, 64, true, true, true><<<g64, blk, 0, stream>>>(bufA, fc4_w, bufB, N);

    // --- Layer 3 (64 -> 128) ---
    gemm16<64, 128, false, false, false><<<g128, blk, 0, stream>>>(bufB, gcn5_w, bufA, N);
    k_selfloop<128><<<cdiv((long long)N * 128, T), blk, 0, stream>>>(bufA, dinv, bufB, N);
    k_scatter <128><<<cdiv((long long)E * 128, T), blk, 0, stream>>>(bufA, src, dst, nrm, bufB, E);
    k_bias_relu<128><<<cdiv((long long)N * 128, T), blk, 0, stream>>>(bufB, gcn5_b, N);
    gemm16<128, 128, true, true, true><<<g128, blk, 0, stream>>>(bufB, fc6_w, out, N);
}